// StelAI1_14525579395104
// MI455X (gfx1250) — compile-verified
//
#include <hip/hip_runtime.h>

typedef float v2f __attribute__((ext_vector_type(2)));
typedef float v8f __attribute__((ext_vector_type(8)));

#define B_    128
#define S_    128
#define H_    1024
#define COMB_ 4096
#define E_    8

// ---------------------------------------------------------------------------
// fp32 WMMA: D = A(16x4) * B(4x16) + C(16x16)
// ---------------------------------------------------------------------------
__device__ __forceinline__ v8f wmma_f32_k4(v2f a, v2f b, v8f c) {
  return __builtin_amdgcn_wmma_f32_16x16x4_f32(
      /*neg_a=*/false, a, /*neg_b=*/false, b,
      /*c_mod=*/(short)0, c, /*reuse_a=*/false, /*reuse_b=*/false);
}

// ---------------------------------------------------------------------------
// CDNA5 async global->LDS copy (ASYNCcnt-tracked, no VGPR round-trip).
// lds_off = LDS byte offset (low 32 bits of a generic LDS pointer).
// ---------------------------------------------------------------------------
__device__ __forceinline__ void async_copy_b64(unsigned lds_off, const float* g) {
  asm volatile("global_load_async_to_lds_b64 %0, %1, off"
               :: "v"(lds_off), "v"(g)
               : "memory");
}
__device__ __forceinline__ void wait_async0() {
  asm volatile("s_wait_asynccnt 0x0" ::: "memory");
}

// ---------------------------------------------------------------------------
// 1) Embedding gather + mean over sequence: pooled[b,h] = mean_s emb[text[b,s],h]
// ---------------------------------------------------------------------------
__global__ void __launch_bounds__(256)
k_embed(const int* __restrict__ text, const float* __restrict__ emb,
        float* __restrict__ pooled) {
  const int b = blockIdx.x;
  const int h = threadIdx.x * 4;             // 256 threads * 4 = 1024 = H
  float4 acc = make_float4(0.f, 0.f, 0.f, 0.f);
  for (int s = 0; s < S_; ++s) {
    const int row = text[b * S_ + s];
    const float4 v = *(const float4*)&emb[(size_t)row * H_ + h];
    acc.x += v.x; acc.y += v.y; acc.z += v.z; acc.w += v.w;
  }
  const float inv = 1.0f / (float)S_;
  float4 o = make_float4(acc.x * inv, acc.y * inv, acc.z * inv, acc.w * inv);
  *(float4*)&pooled[(size_t)b * H_ + h] = o;
}

// ---------------------------------------------------------------------------
// 2) Conv3x3 SAME + bias + relu + global average pool, into x[:, H..2H)
// ---------------------------------------------------------------------------
__global__ void __launch_bounds__(256)
k_conv(const float* __restrict__ img, const float* __restrict__ cW,
       const float* __restrict__ cb, float* __restrict__ xout,
       int ld_out, int ocol) {
  __shared__ float simg[3 * 32 * 32];
  const int b   = blockIdx.x;
  const int tid = threadIdx.x;
  for (int i = tid; i < 3072; i += 256) simg[i] = img[(size_t)b * 3072 + i];
  __syncthreads();

  const int o  = blockIdx.y * 64 + (tid >> 2);
  const int pg = tid & 3;
  float wr[27];
#pragma unroll
  for (int i = 0; i < 27; ++i) wr[i] = cW[o * 27 + i];
  const float bias = cb[o];

  float accum = 0.f;
  for (int p = pg * 256; p < pg * 256 + 256; ++p) {
    const int y  = p >> 5;
    const int xx = p & 31;
    float s = bias;
#pragma unroll
    for (int c = 0; c < 3; ++c) {
#pragma unroll
      for (int ky = 0; ky < 3; ++ky) {
        const int yy = y + ky - 1;
        if ((unsigned)yy < 32u) {
#pragma unroll
          for (int kx = 0; kx < 3; ++kx) {
            const int xr = xx + kx - 1;
            if ((unsigned)xr < 32u)
              s += wr[(c * 3 + ky) * 3 + kx] * simg[c * 1024 + yy * 32 + xr];
          }
        }
      }
    }
    accum += fmaxf(s, 0.f);
  }
  accum += __shfl_down(accum, 2, 4);
  accum += __shfl_down(accum, 1, 4);
  if (pg == 0)
    xout[(size_t)b * ld_out + ocol + o] = accum * (1.0f / 1024.0f);
}

// ---------------------------------------------------------------------------
// 3) WMMA GEMM, async double-buffered weight staging.
//    out[m, ocol+col] = act(sum_k A[m,k]*W[k,col] + bias[col])
//    A: [128, K] row-major. W: [K, ldW] row-major. grid.x = 64 (N strips of 16).
// ---------------------------------------------------------------------------
__global__ void __launch_bounds__(256)
k_gemm(const float* __restrict__ A, const float* __restrict__ W,
       const float* __restrict__ bias, float* __restrict__ out,
       int K, int ldW, int ld_out, int ocol, int do_relu) {
  __shared__ float sW[2][32 * 16];
  const int tid   = threadIdx.x;
  const int wave  = tid >> 5;
  const int lane  = tid & 31;
  const int lmod  = lane & 15;
  const int lhalf = lane >> 4;
  const int koff  = lhalf * 2;            // f32 WMMA A/B: lanes 16..31 hold K=2,3
  const int col0  = blockIdx.x * 16;
  const int arow  = wave * 16 + lmod;
  const int sr = (tid * 2) >> 4;          // staged row within 32xK chunk
  const int sc = (tid * 2) & 15;          // staged col (even)

  const unsigned lwr[2] = { (unsigned)(size_t)&sW[0][tid * 2],
                            (unsigned)(size_t)&sW[1][tid * 2] };

  // prologue: async-stage chunk 0 into buffer 0
  async_copy_b64(lwr[0], &W[(size_t)sr * ldW + col0 + sc]);
  wait_async0();
  __syncthreads();

  v8f acc = {};
  int buf = 0;
  for (int k0 = 0; k0 < K; k0 += 32, buf ^= 1) {
    if (k0 + 32 < K)  // overlap: stage chunk k+1 while computing chunk k
      async_copy_b64(lwr[buf ^ 1],
                     &W[(size_t)(k0 + 32 + sr) * ldW + col0 + sc]);
    const float* ap = &A[(size_t)arow * K + k0 + koff];
    __builtin_prefetch(ap + 64, 0, 1);    // gfx1250 global_prefetch_b8 (A stream)
    const float* sb = &sW[buf][0];
#pragma unroll
    for (int kk = 0; kk < 32; kk += 4) {
      v2f a, b;
      a.x = ap[kk];
      a.y = ap[kk + 1];
      b.x = sb[(kk + koff) * 16 + lmod];
      b.y = sb[(kk + koff + 1) * 16 + lmod];
      acc = wmma_f32_k4(a, b, acc);
    }
    wait_async0();
    __syncthreads();
  }

  const float bv   = bias[col0 + lmod];
  const int  rbase = wave * 16 + lhalf * 8;
#pragma unroll
  for (int r = 0; r < 8; ++r) {
    float v = acc[r] + bv;
    if (do_relu) v = fmaxf(v, 0.f);
    out[(size_t)(rbase + r) * ld_out + ocol + col0 + lmod] = v;
  }
}

// ---------------------------------------------------------------------------
// 4) Gate: logits = x @ gate_W + b -> softmax -> top-2 sparse weights [B, 8]
// ---------------------------------------------------------------------------
__global__ void __launch_bounds__(256)
k_gate(const float* __restrict__ X, const float* __restrict__ gW,
       const float* __restrict__ gb, float* __restrict__ wgt) {
  __shared__ float red[E_][256];
  const int b   = blockIdx.x;
  const int tid = threadIdx.x;
  float acc[E_];
#pragma unroll
  for (int e = 0; e < E_; ++e) acc[e] = 0.f;
  for (int d = tid; d < COMB_; d += 256) {
    const float xv = X[(size_t)b * COMB_ + d];
#pragma unroll
    for (int e = 0; e < E_; ++e) acc[e] += xv * gW[d * E_ + e];
  }
#pragma unroll
  for (int e = 0; e < E_; ++e) red[e][tid] = acc[e];
  __syncthreads();
  for (int off = 128; off > 0; off >>= 1) {
    if (tid < off) {
#pragma unroll
      for (int e = 0; e < E_; ++e) red[e][tid] += red[e][tid + off];
    }
    __syncthreads();
  }
  if (tid == 0) {
    float lg[E_], m = -1e30f;
#pragma unroll
    for (int e = 0; e < E_; ++e) { lg[e] = red[e][0] + gb[e]; m = fmaxf(m, lg[e]); }
    float p[E_], sum = 0.f;
#pragma unroll
    for (int e = 0; e < E_; ++e) { p[e] = __expf(lg[e] - m); sum += p[e]; }
    const float inv = 1.f / sum;
#pragma unroll
    for (int e = 0; e < E_; ++e) p[e] *= inv;
    int i1 = 0;
#pragma unroll
    for (int e = 1; e < E_; ++e) if (p[e] > p[i1]) i1 = e;
    int i2 = (i1 == 0) ? 1 : 0;
#pragma unroll
    for (int e = 0; e < E_; ++e)
      if (e != i1 && e != i2 && p[e] > p[i2]) i2 = e;
#pragma unroll
    for (int e = 0; e < E_; ++e)
      wgt[b * E_ + e] = (e == i1) ? p[i1] : (e == i2) ? p[i2] : 0.f;
  }
}

// ---------------------------------------------------------------------------
// 5) MoE: moe[b,h] = sum_e w[b,e] * (x[b,:] @ expert_W[e,:,h] + expert_b[e,h])
//    Async double-buffered weight staging; weighted combine fused in registers.
// ---------------------------------------------------------------------------
__global__ void __launch_bounds__(256)
k_moe(const float* __restrict__ X, const float* __restrict__ eW,
      const float* __restrict__ eb, const float* __restrict__ wgt,
      float* __restrict__ moe) {
  __shared__ float sW[2][32 * 16];
  const int tid   = threadIdx.x;
  const int wave  = tid >> 5;
  const int lane  = tid & 31;
  const int lmod  = lane & 15;
  const int lhalf = lane >> 4;
  const int koff  = lhalf * 2;
  const int col0  = blockIdx.x * 16;
  const int arow  = wave * 16 + lmod;
  const int sr = (tid * 2) >> 4;
  const int sc = (tid * 2) & 15;
  const int rbase = wave * 16 + lhalf * 8;

  const unsigned lwr[2] = { (unsigned)(size_t)&sW[0][tid * 2],
                            (unsigned)(size_t)&sW[1][tid * 2] };

  // prologue: stage expert 0 / chunk 0
  async_copy_b64(lwr[0], &eW[(size_t)sr * H_ + col0 + sc]);
  wait_async0();
  __syncthreads();

  v8f macc = {};
  int buf = 0;
  for (int e = 0; e < E_; ++e) {
    const float* We = eW + (size_t)e * COMB_ * H_;
    v8f acc = {};
    for (int k0 = 0; k0 < COMB_; k0 += 32, buf ^= 1) {
      // stage next chunk: either k0+32 of this expert, or chunk 0 of expert e+1
      if (k0 + 32 < COMB_) {
        async_copy_b64(lwr[buf ^ 1],
                       &We[(size_t)(k0 + 32 + sr) * H_ + col0 + sc]);
      } else if (e + 1 < E_) {
        async_copy_b64(lwr[buf ^ 1],
                       &We[(size_t)(COMB_ + sr) * H_ + col0 + sc]); // == expert e+1, k=sr
      }
      const float* ap = &X[(size_t)arow * COMB_ + k0 + koff];
      __builtin_prefetch(ap + 64, 0, 1);
      const float* sb = &sW[buf][0];
#pragma unroll
      for (int kk = 0; kk < 32; kk += 4) {
        v2f a, b;
        a.x = ap[kk];
        a.y = ap[kk + 1];
        b.x = sb[(kk + koff) * 16 + lmod];
        b.y = sb[(kk + koff + 1) * 16 + lmod];
        acc = wmma_f32_k4(a, b, acc);
      }
      wait_async0();
      __syncthreads();
    }
    const float bv = eb[e * H_ + col0 + lmod];
#pragma unroll
    for (int r = 0; r < 8; ++r)
      macc[r] += wgt[(rbase + r) * E_ + e] * (acc[r] + bv);
  }
#pragma unroll
  for (int r = 0; r < 8; ++r)
    moe[(size_t)(rbase + r) * H_ + col0 + lmod] = macc[r];
}

// ---------------------------------------------------------------------------
extern "C" void kernel_launch(void* const* d_in, const int* in_sizes, int n_in,
                              void* d_out, int out_size, void* d_ws, size_t ws_size,
                              hipStream_t stream) {
  (void)in_sizes; (void)n_in; (void)out_size; (void)ws_size;
  const int*   text    = (const int*)  d_in[0];
  const float* image   = (const float*)d_in[1];
  const float* audio   = (const float*)d_in[2];
  const float* video   = (const float*)d_in[3];
  const float* emb     = (const float*)d_in[4];
  const float* text_W  = (const float*)d_in[5];
  const float* text_b  = (const float*)d_in[6];
  const float* conv_W  = (const float*)d_in[7];
  const float* conv_b  = (const float*)d_in[8];
  const float* audio_W = (const float*)d_in[9];
  const float* audio_b = (const float*)d_in[10];
  const float* video_W = (const float*)d_in[11];
  const float* video_b = (const float*)d_in[12];
  const float* exp_W   = (const float*)d_in[13];
  const float* exp_b   = (const float*)d_in[14];
  const float* gate_W  = (const float*)d_in[15];
  const float* gate_b  = (const float*)d_in[16];
  const float* rsn_W   = (const float*)d_in[17];
  const float* rsn_b   = (const float*)d_in[18];

  float* pooled = (float*)d_ws;                 // [128, 1024]
  float* x      = pooled + B_ * H_;             // [128, 4096] concat buffer
  float* wgt    = x + B_ * COMB_;               // [128, 8]
  float* moe    = wgt + B_ * E_;                // [128, 1024]
  float* out    = (float*)d_out;                // [128, 1024]

  k_embed<<<B_, 256, 0, stream>>>(text, emb, pooled);
  k_conv <<<dim3(B_, 16), 256, 0, stream>>>(image, conv_W, conv_b, x, COMB_, H_);
  k_gemm <<<H_ / 16, 256, 0, stream>>>(pooled, text_W,  text_b,  x,   H_,    H_, COMB_, 0,      1);
  k_gemm <<<H_ / 16, 256, 0, stream>>>(audio,  audio_W, audio_b, x,   2048,  H_, COMB_, 2 * H_, 1);
  k_gemm <<<H_ / 16, 256, 0, stream>>>(video,  video_W, video_b, x,   4096,  H_, COMB_, 3 * H_, 1);
  k_gate <<<B_, 256, 0, stream>>>(x, gate_W, gate_b, wgt);
  k_moe  <<<H_ / 16, 256, 0, stream>>>(x, exp_W, exp_b, wgt, moe);
  k_gemm <<<H_ / 16, 256, 0, stream>>>(moe, rsn_W, rsn_b, out, H_, H_, H_, 0, 0);
}